// OneToOneLinear_24807731102291
// MI455X (gfx1250) — compile-verified
//
#include <hip/hip_runtime.h>

// sigmoid(SCALE * (x * w + b)) over a 32768 x 2048 f32 matrix.
// Pure streaming / bandwidth-bound: 512 MB of mandatory HBM traffic,
// ~22 us floor at 23.3 TB/s. No reduction -> WMMA inapplicable.
// Strategy: b128 NT loads/stores, per-thread hoisted w/b, global_prefetch.

#define FEATURES 2048
#define NROWS    32768
#define SCALE    4.0f

typedef float f32x4 __attribute__((ext_vector_type(4)));

__global__ __launch_bounds__(256) void
OneToOneLinear_sigscale_kernel(const float* __restrict__ in,
                               const float* __restrict__ weight,
                               const float* __restrict__ bias,
                               float* __restrict__ out)
{
    // Each thread owns 4 consecutive features (one float4 column slot).
    // gridDim.x * blockDim.x * 4 == FEATURES  (2 blocks x 256 threads x 4).
    const int col4 = blockIdx.x * blockDim.x + threadIdx.x;  // 0..511
    const int col  = col4 * 4;

    // Hoist per-feature constants: fold SCALE into both.
    // 8 KB arrays -> stay hot in L1/L2; loaded once per thread.
    f32x4 wv = *(const f32x4*)(weight + col);
    f32x4 bv = *(const f32x4*)(bias   + col);
    wv *= SCALE;
    bv *= SCALE;

    const size_t stride = (size_t)gridDim.y * FEATURES;   // row stride in floats
    const float* ip = in  + (size_t)blockIdx.y * FEATURES + col;
    float*       op = out + (size_t)blockIdx.y * FEATURES + col;

    const float NLOG2E = -1.44269504088896340736f;

    for (int r = blockIdx.y; r < NROWS; r += gridDim.y) {
        // Speculative streaming prefetch a few rows ahead (global_prefetch_b8).
        // Speculative RT prefetch: invalid tail addresses silently dropped.
        __builtin_prefetch(ip + 4 * stride, 0, 0);

        // Non-temporal 128-bit load: data touched exactly once, working set
        // (512 MB) exceeds L2 (192 MB) -> don't pollute caches.
        f32x4 x = __builtin_nontemporal_load((const f32x4*)ip);

        f32x4 y;
#pragma unroll
        for (int k = 0; k < 4; ++k) {
            float t = __builtin_fmaf(wv[k], x[k], bv[k]);     // SCALE*(x*w+b)
            float e = __builtin_amdgcn_exp2f(t * NLOG2E);     // exp(-t) via v_exp_f32
            y[k] = 1.0f / (1.0f + e);                         // IEEE divide for accuracy
        }

        // Non-temporal 128-bit store of the output stream.
        __builtin_nontemporal_store(y, (f32x4*)op);

        ip += stride;
        op += stride;
    }
}

extern "C" void kernel_launch(void* const* d_in, const int* in_sizes, int n_in,
                              void* d_out, int out_size, void* d_ws, size_t ws_size,
                              hipStream_t stream)
{
    const float* in     = (const float*)d_in[0];  // (32768, 2048) f32
    const float* weight = (const float*)d_in[1];  // (2048,) f32
    const float* bias   = (const float*)d_in[2];  // (2048,) f32
    float*       out    = (float*)d_out;          // (32768, 2048) f32

    (void)in_sizes; (void)n_in; (void)out_size; (void)d_ws; (void)ws_size;

    // grid.x covers the 2048 features (2 blocks x 256 threads x float4),
    // grid.y strides the 32768 rows (8 rows per thread) -> 8192 blocks,
    // 64K wave32 waves: plenty of memory-level parallelism for 23.3 TB/s.
    dim3 block(256, 1, 1);
    dim3 grid(FEATURES / (4 * 256), 4096, 1);
    OneToOneLinear_sigscale_kernel<<<grid, block, 0, stream>>>(in, weight, bias, out);
}